// VisionMamba_65429531788054
// MI455X (gfx1250) — compile-verified
//
#include <hip/hip_runtime.h>
#include <hip/hip_bf16.h>

// ---------------- model constants ----------------
#define BB      16
#define IMG     224
#define PP      16
#define CIN     3
#define DM      192
#define DEPTH   24
#define DI      384
#define DS      16
#define DCONV   4
#define DTR     12
#define NCLS    1000
#define NPATCH  196          // (224/16)^2
#define LL      197          // NPATCH + 1 (cls at index 98)
#define TP      98           // cls position
#define EPSF    1e-5f

typedef __attribute__((ext_vector_type(2))) float v2f;
typedef __attribute__((ext_vector_type(8))) float v8f;

// =====================================================================
// WMMA f32 GEMM:  Dst[M,N] = A[M,K] * W[N,K]^T    (all row-major, fp32)
// Register-blocked: one wave computes a 16x64 strip of Dst (4 N-tiles),
// reusing the A fragment across 4 V_WMMA_F32_16X16X4_F32 per K-step.
// M multiple of 16 (grid-exact), N multiple of 64, K multiple of 4.
// EXEC stays all-ones inside every live wave (WMMA requirement).
//
// Wave32 operand layout (ISA 7.12.2):
//   A (16x4):  lanes 0-15 -> M=lane, K={k,k+1}; lanes 16-31 -> K={k+2,k+3}
//   B (4x16):  lanes 0-15 -> N=lane, K={k,k+1}; lanes 16-31 -> K={k+2,k+3}
//   C/D:       col = lane%16, VGPR j -> row j (lanes<16) / row j+8 (lanes>=16)
// =====================================================================
template<int K>
__global__ void gemm_wmma_f32(const float* __restrict__ A,
                              const float* __restrict__ W,
                              float* __restrict__ Dst,
                              int N) {
    const int lane = threadIdx.x & 31;
    const int wave = threadIdx.x >> 5;
    const int wpb  = blockDim.x >> 5;
    const int m0   = blockIdx.x * 16;
    const int n0   = (blockIdx.y * wpb + wave) * 64;   // 4 N-tiles per wave
    if (n0 >= N) return;                 // wave-uniform: EXEC all-ones below

    const int lm = lane & 15;            // row (A) / col (B) within tile
    const int hi = lane >> 4;            // 0: K pair {0,1}; 1: K pair {2,3}

    const float* __restrict__ arow = A + (size_t)(m0 + lm) * K + 2 * hi;
    const float* __restrict__ wrow = W + (size_t)(n0 + lm) * K + 2 * hi;

    v8f acc0 = {}, acc1 = {}, acc2 = {}, acc3 = {};
#pragma unroll 2
    for (int k = 0; k < K; k += 4) {
        v2f a, b0, b1, b2, b3;
        a.x  = arow[k];               a.y  = arow[k + 1];
        b0.x = wrow[k];               b0.y = wrow[k + 1];
        b1.x = wrow[k + 16 * K];      b1.y = wrow[k + 16 * K + 1];
        b2.x = wrow[k + 32 * K];      b2.y = wrow[k + 32 * K + 1];
        b3.x = wrow[k + 48 * K];      b3.y = wrow[k + 48 * K + 1];
        acc0 = __builtin_amdgcn_wmma_f32_16x16x4_f32(false, a, false, b0,
                                                     (short)0, acc0, false, false);
        acc1 = __builtin_amdgcn_wmma_f32_16x16x4_f32(false, a, false, b1,
                                                     (short)0, acc1, false, false);
        acc2 = __builtin_amdgcn_wmma_f32_16x16x4_f32(false, a, false, b2,
                                                     (short)0, acc2, false, false);
        acc3 = __builtin_amdgcn_wmma_f32_16x16x4_f32(false, a, false, b3,
                                                     (short)0, acc3, false, false);
    }

    float* __restrict__ drow = Dst + (size_t)(m0 + hi * 8) * N + n0 + lm;
#pragma unroll
    for (int j = 0; j < 8; ++j) {
        drow[(size_t)j * N + 0]  = acc0[j];
        drow[(size_t)j * N + 16] = acc1[j];
        drow[(size_t)j * N + 32] = acc2[j];
        drow[(size_t)j * N + 48] = acc3[j];
    }
}

// =====================================================================
// im2col for the 16x16/stride-16 patch conv: col[b*196+p][c*256+i*16+j]
// =====================================================================
__global__ void im2col_kernel(const float* __restrict__ x, float* __restrict__ col) {
    int idx = blockIdx.x * blockDim.x + threadIdx.x;
    const int total = BB * NPATCH * (CIN * PP * PP);
    if (idx >= total) return;
    int cc  = idx % (CIN * PP * PP);
    int row = idx / (CIN * PP * PP);
    int b   = row / NPATCH;
    int p   = row % NPATCH;
    int py  = p / (IMG / PP);
    int px  = p % (IMG / PP);
    int c   = cc / (PP * PP);
    int r   = cc % (PP * PP);
    int i   = r / PP;
    int j   = r % PP;
    col[idx] = x[(((size_t)b * CIN + c) * IMG + (py * PP + i)) * IMG + (px * PP + j)];
}

// =====================================================================
// assemble tokens: insert cls at TP, add patch bias + pos embed, zero res
// =====================================================================
__global__ void assemble_kernel(const float* __restrict__ xe,
                                const float* __restrict__ pb,
                                const float* __restrict__ cls,
                                const float* __restrict__ pos,
                                float* __restrict__ h,
                                float* __restrict__ res) {
    int idx = blockIdx.x * blockDim.x + threadIdx.x;
    const int total = BB * LL * DM;
    if (idx >= total) return;
    int d  = idx % DM;
    int bl = idx / DM;
    int l  = bl % LL;
    int b  = bl / LL;
    float v;
    if (l == TP) {
        v = cls[d];
    } else {
        int p = (l < TP) ? l : l - 1;
        v = xe[((size_t)b * NPATCH + p) * DM + d] + pb[d];
    }
    h[idx]   = v + pos[l * DM + d];
    res[idx] = 0.f;
}

// =====================================================================
// fused add + RMSNorm, one wave32 per token (6 elems/lane)
// res <- h + res ;  hn <- (h+res) * rsqrt(mean^2) * w
// =====================================================================
__global__ void rmsnorm_kernel(const float* __restrict__ h,
                               float* __restrict__ res,
                               float* __restrict__ hn,
                               const float* __restrict__ w) {
    const int tok  = blockIdx.x;           // 0..B*L-1
    const int lane = threadIdx.x;          // 32 threads
    const float* hp = h   + (size_t)tok * DM;
    float*       rp = res + (size_t)tok * DM;
    float xv[DM / 32];
    float ss = 0.f;
#pragma unroll
    for (int i = 0; i < DM / 32; ++i) {
        int d = lane + i * 32;
        float v = hp[d] + rp[d];
        xv[i] = v;
        ss += v * v;
    }
#pragma unroll
    for (int m = 16; m >= 1; m >>= 1)
        ss += __shfl_xor(ss, m, 32);
    float rstd = rsqrtf(ss / (float)DM + EPSF);
#pragma unroll
    for (int i = 0; i < DM / 32; ++i) {
        int d = lane + i * 32;
        rp[d]                     = xv[i];
        hn[(size_t)tok * DM + d]  = xv[i] * rstd * w[d];
    }
}

// =====================================================================
// causal depthwise conv1d (k=4, left pad 3) + bias + SiLU.
// xm lives in xz[..., 0:DI]; output dense xact[b,l,c].
// =====================================================================
__global__ void conv_silu_kernel(const float* __restrict__ xz,
                                 const float* __restrict__ cw,
                                 const float* __restrict__ cb,
                                 float* __restrict__ xact) {
    int idx = blockIdx.x * blockDim.x + threadIdx.x;
    const int total = BB * LL * DI;
    if (idx >= total) return;
    int c  = idx % DI;
    int bl = idx / DI;
    int l  = bl % LL;
    int b  = bl / LL;
    float s = cb[c];
#pragma unroll
    for (int k = 0; k < DCONV; ++k) {
        int ls = l - (DCONV - 1) + k;
        if (ls >= 0)
            s += xz[((size_t)(b * LL + ls)) * (2 * DI) + c] * cw[c * DCONV + k];
    }
    xact[idx] = s / (1.f + __expf(-s));    // silu
}

// =====================================================================
// x_proj (N=44) + dt_proj (r=12) + softplus; one 384-thread block/token.
// Emits dt[b,l,d], Bm[b,l,s], Cm[b,l,s].
// =====================================================================
__global__ void xproj_dt_kernel(const float* __restrict__ xact,
                                const float* __restrict__ xpw,   // [44,384]
                                const float* __restrict__ dtw,   // [384,12]
                                const float* __restrict__ dtb,   // [384]
                                float* __restrict__ dt,
                                float* __restrict__ Bm,
                                float* __restrict__ Cm) {
    __shared__ float xl[DI];
    __shared__ float dbl[DTR + 2 * DS];
    const int tok = blockIdx.x;
    const int t   = threadIdx.x;           // 0..383
    xl[t] = xact[(size_t)tok * DI + t];
    __syncthreads();
    if (t < DTR + 2 * DS) {
        const float* wr = xpw + t * DI;
        float s = 0.f;
        for (int i = 0; i < DI; ++i) s += xl[i] * wr[i];
        dbl[t] = s;
    }
    __syncthreads();
    {   // dt for channel d = t
        const float* wr = dtw + t * DTR;
        float s = dtb[t];
#pragma unroll
        for (int r = 0; r < DTR; ++r) s += dbl[r] * wr[r];
        dt[(size_t)tok * DI + t] = (s > 20.f) ? s : log1pf(__expf(s));  // softplus
    }
    if (t < DS)            Bm[(size_t)tok * DS + t]        = dbl[DTR + t];
    else if (t < 2 * DS)   Cm[(size_t)tok * DS + (t - DS)] = dbl[DTR + t];
}

// =====================================================================
// selective scan over L: 16 lanes (states) per (b,d) pair; intra-wave
// shuffle reduction for y = sum_s h_s * C_s, plus x*D skip.
// =====================================================================
__global__ void scan_kernel(const float* __restrict__ dt,
                            const float* __restrict__ xact,
                            const float* __restrict__ Bm,
                            const float* __restrict__ Cm,
                            const float* __restrict__ A_log,  // [384,16]
                            const float* __restrict__ Dp,     // [384]
                            float* __restrict__ y) {
    const int t    = threadIdx.x;
    const int s    = t & 15;                    // state index
    const int g    = t >> 4;                    // group within block
    const int pair = blockIdx.x * (blockDim.x >> 4) + g;   // 0..B*DI-1
    const int b    = pair / DI;
    const int d    = pair % DI;

    const float a  = -__expf(A_log[d * DS + s]);
    const float Dv = Dp[d];
    float hst = 0.f;
    for (int l = 0; l < LL; ++l) {
        const size_t td = ((size_t)(b * LL + l)) * DI + d;
        const size_t ts = ((size_t)(b * LL + l)) * DS + s;
        float dtv = dt[td];
        float xv  = xact[td];
        float Bv  = Bm[ts];
        float Cv  = Cm[ts];
        hst = __expf(dtv * a) * hst + dtv * Bv * xv;
        float c = hst * Cv;
        c += __shfl_xor(c, 1, 32);
        c += __shfl_xor(c, 2, 32);
        c += __shfl_xor(c, 4, 32);
        c += __shfl_xor(c, 8, 32);
        if (s == 0) y[td] = c + xv * Dv;
    }
}

// =====================================================================
// gate: y <- y * silu(z), z = xz[..., DI:2*DI]
// =====================================================================
__global__ void gate_kernel(float* __restrict__ y, const float* __restrict__ xz) {
    int idx = blockIdx.x * blockDim.x + threadIdx.x;
    const int total = BB * LL * DI;
    if (idx >= total) return;
    int c  = idx % DI;
    int bl = idx / DI;
    float zv = xz[(size_t)bl * (2 * DI) + DI + c];
    y[idx] = y[idx] * (zv / (1.f + __expf(-zv)));
}

// =====================================================================
// final: fused add+RMSNorm at the cls token + classifier head
// =====================================================================
__global__ void final_head_kernel(const float* __restrict__ h,
                                  const float* __restrict__ res,
                                  const float* __restrict__ nw,
                                  const float* __restrict__ hw,   // [1000,192]
                                  const float* __restrict__ hb,
                                  float* __restrict__ out) {
    __shared__ float xn[DM];
    __shared__ float red[256];
    __shared__ float rstd_s;
    const int b = blockIdx.x;
    const int t = threadIdx.x;
    float v = 0.f;
    if (t < DM) {
        size_t o = ((size_t)(b * LL + TP)) * DM + t;
        v = h[o] + res[o];
        xn[t] = v;
    }
    red[t] = v * v;
    __syncthreads();
    for (int st = 128; st > 0; st >>= 1) {
        if (t < st) red[t] += red[t + st];
        __syncthreads();
    }
    if (t == 0) rstd_s = rsqrtf(red[0] / (float)DM + EPSF);
    __syncthreads();
    if (t < DM) xn[t] = xn[t] * rstd_s * nw[t];
    __syncthreads();
    for (int n = t; n < NCLS; n += 256) {
        const float* wr = hw + (size_t)n * DM;
        float s = hb[n];
        for (int i = 0; i < DM; ++i) s += xn[i] * wr[i];
        out[(size_t)b * NCLS + n] = s;
    }
}

// =====================================================================
// host driver
// =====================================================================
extern "C" void kernel_launch(void* const* d_in, const int* in_sizes, int n_in,
                              void* d_out, int out_size, void* d_ws, size_t ws_size,
                              hipStream_t stream) {
    const float* x          = (const float*)d_in[0];
    const float* patch_w    = (const float*)d_in[1];
    const float* patch_b    = (const float*)d_in[2];
    const float* cls_token  = (const float*)d_in[3];
    const float* pos_embed  = (const float*)d_in[4];
    const float* norm_w     = (const float*)d_in[5];
    const float* in_proj_w  = (const float*)d_in[6];
    const float* conv_w     = (const float*)d_in[7];
    const float* conv_b     = (const float*)d_in[8];
    const float* x_proj_w   = (const float*)d_in[9];
    const float* dt_proj_w  = (const float*)d_in[10];
    const float* dt_proj_b  = (const float*)d_in[11];
    const float* A_log      = (const float*)d_in[12];
    const float* Dparam     = (const float*)d_in[13];
    const float* out_proj_w = (const float*)d_in[14];
    const float* norm_f_w   = (const float*)d_in[15];
    const float* head_w     = (const float*)d_in[16];
    const float* head_b     = (const float*)d_in[17];

    float* ws = (float*)d_ws;
    size_t off = 0;
    auto carve = [&](size_t n) { float* p = ws + off; off += n; return p; };

    float* col  = carve((size_t)BB * NPATCH * CIN * PP * PP);  // im2col  [3136,768]
    float* xe   = carve((size_t)BB * NPATCH * DM);             // patch out [3136,192]
    float* h    = carve((size_t)BB * LL * DM);
    float* res  = carve((size_t)BB * LL * DM);
    float* hn   = carve((size_t)BB * LL * DM);
    float* xz   = carve((size_t)BB * LL * 2 * DI);             // [3152,768]
    float* xact = carve((size_t)BB * LL * DI);
    float* dtb_ = carve((size_t)BB * LL * DI);
    float* Bm   = carve((size_t)BB * LL * DS);
    float* Cm   = carve((size_t)BB * LL * DS);
    float* ybuf = carve((size_t)BB * LL * DI);
    (void)ws_size;

    const int TOK = BB * LL;            // 3152 = 197*16

    // ---- prologue: patch embed (im2col + WMMA GEMM) + token assembly ----
    {
        int n = BB * NPATCH * CIN * PP * PP;
        im2col_kernel<<<(n + 255) / 256, 256, 0, stream>>>(x, col);
        // M tiles = 3136/16 = 196; N strips = 192/64 = 3 -> 1 block of 4 waves (1 idle)
        dim3 gp(NPATCH, 1);
        gemm_wmma_f32<CIN * PP * PP><<<gp, 128, 0, stream>>>(col, patch_w, xe, DM);
        int m = BB * LL * DM;
        assemble_kernel<<<(m + 255) / 256, 256, 0, stream>>>(xe, patch_b, cls_token,
                                                             pos_embed, h, res);
    }

    // ---- 24 mamba layers ----
    for (int layer = 0; layer < DEPTH; ++layer) {
        rmsnorm_kernel<<<TOK, 32, 0, stream>>>(h, res, hn, norm_w + layer * DM);

        // in_proj: M=3152 -> 197 tiles; N=768 -> 12 strips of 64 -> grid.y = 3
        dim3 g1(LL, 3);
        gemm_wmma_f32<DM><<<g1, 128, 0, stream>>>(hn, in_proj_w + (size_t)layer * 2 * DI * DM,
                                                  xz, 2 * DI);

        int ne = BB * LL * DI;
        conv_silu_kernel<<<(ne + 255) / 256, 256, 0, stream>>>(
            xz, conv_w + (size_t)layer * DI * DCONV, conv_b + (size_t)layer * DI, xact);

        xproj_dt_kernel<<<TOK, DI, 0, stream>>>(
            xact, x_proj_w + (size_t)layer * (DTR + 2 * DS) * DI,
            dt_proj_w + (size_t)layer * DI * DTR, dt_proj_b + (size_t)layer * DI,
            dtb_, Bm, Cm);

        scan_kernel<<<(BB * DI) / 16, 256, 0, stream>>>(
            dtb_, xact, Bm, Cm, A_log + (size_t)layer * DI * DS,
            Dparam + (size_t)layer * DI, ybuf);

        gate_kernel<<<(ne + 255) / 256, 256, 0, stream>>>(ybuf, xz);

        // out_proj: N=192 -> 3 strips of 64 -> grid.y = 1 (1 idle wave per block)
        dim3 g2(LL, 1);
        gemm_wmma_f32<DI><<<g2, 128, 0, stream>>>(ybuf, out_proj_w + (size_t)layer * DM * DI,
                                                  h, DM);
    }

    // ---- final norm + head ----
    final_head_kernel<<<BB, 256, 0, stream>>>(h, res, norm_f_w, head_w, head_b,
                                              (float*)d_out);
}